// CoarseGrain_26474178413228
// MI455X (gfx1250) — compile-verified
//
#include <hip/hip_runtime.h>

// CoarseGrain: avg-pool window=4 over last axis of (128,4,65536) fp32.
// Flat: out[o] = 0.25f * (in[4o] + in[4o+1] + in[4o+2] + in[4o+3]).
//
// Memory-bound: 160 MB traffic/call -> ~6.9 us HBM floor @ 23.3 TB/s.
// Working set fits in the 192 MB L2, so default (RT) cache policy is used
// deliberately: graph replays can then serve the input from L2.

typedef float v4f __attribute__((ext_vector_type(4)));

// Each thread: 4 x b128 loads (4 windows), 1 x b128 store (4 outputs).
__global__ __launch_bounds__(256) void coarse_grain_pool4_vec(
    const v4f* __restrict__ in, v4f* __restrict__ out, int n4 /* #float4 outputs */) {
  int t = blockIdx.x * blockDim.x + threadIdx.x;
  if (t >= n4) return;

  const v4f* p = in + ((long)t << 2);   // 4 input float4s per thread
  v4f a0 = p[0];
  v4f a1 = p[1];
  v4f a2 = p[2];
  v4f a3 = p[3];

  v4f r;
  r.x = ((a0.x + a0.y) + (a0.z + a0.w)) * 0.25f;
  r.y = ((a1.x + a1.y) + (a1.z + a1.w)) * 0.25f;
  r.z = ((a2.x + a2.y) + (a2.z + a2.w)) * 0.25f;
  r.w = ((a3.x + a3.y) + (a3.z + a3.w)) * 0.25f;

  out[t] = r;
}

// Scalar tail (only used if out_size % 4 != 0; not hit for the given shapes).
__global__ __launch_bounds__(256) void coarse_grain_pool4_tail(
    const float* __restrict__ in, float* __restrict__ out, int start, int n) {
  int o = start + blockIdx.x * blockDim.x + threadIdx.x;
  if (o >= n) return;
  const float* p = in + 4L * o;
  out[o] = ((p[0] + p[1]) + (p[2] + p[3])) * 0.25f;
}

extern "C" void kernel_launch(void* const* d_in, const int* in_sizes, int n_in,
                              void* d_out, int out_size, void* d_ws, size_t ws_size,
                              hipStream_t stream) {
  (void)in_sizes; (void)n_in; (void)d_ws; (void)ws_size;

  const float* x = (const float*)d_in[0];
  float* y = (float*)d_out;

  int n4 = out_size >> 2;              // 2,097,152 for the given shapes
  if (n4 > 0) {
    int block = 256;
    int grid = (n4 + block - 1) / block;
    coarse_grain_pool4_vec<<<grid, block, 0, stream>>>(
        (const v4f*)x, (v4f*)y, n4);
  }

  int done = n4 << 2;
  int rem = out_size - done;
  if (rem > 0) {
    coarse_grain_pool4_tail<<<1, 256, 0, stream>>>(x, y, done, out_size);
  }
}